// SpecialConvBlock_2516850835704
// MI455X (gfx1250) — compile-verified
//
#include <hip/hip_runtime.h>
#include <math.h>

typedef float v2f __attribute__((ext_vector_type(2)));
typedef float v8f __attribute__((ext_vector_type(8)));

#define N_IMG 16
#define NBR   8          // branches
#define CIN   3
#define HIN   224
#define WIN   224
#define COUT  64
#define OHC   112        // conv out H
#define OWC   112        // conv out W
#define PHO   56         // pool out H
#define PWO   56         // pool out W
#define KDIM  147        // CIN*7*7
#define WSH_STRIDE 65    // padded LDS stride for [k][co] weights
#define WSH_ROWS   148   // 147 + 1 zero row for K remainder
#define WSH_FLOATS (WSH_ROWS * WSH_STRIDE)        // 9620
#define PATCH_W    37    // input cols covered by a 16-px tile: 2*15+6+1
#define PATCH_H    9     // input rows covering output rows oh and oh+1
#define PATCH_FLOATS (CIN * PATCH_H * PATCH_W)    // 999
#define ROW1_OFF   (2 * PATCH_W)                  // +2 input rows for tile 1
#define NWAVES     4
#define TR_STRIDE  18    // transpose row stride (8B-aligned pairs)
#define TR_FLOATS  (2 * COUT * TR_STRIDE)         // 2304 per wave (2 tiles)
#define TILES_PER_BRANCH (N_IMG * (OHC/2) * (OWC/16)) // 16*56*7 = 6272
#define BLOCKS_PER_BRANCH (TILES_PER_BRANCH / NWAVES) // 1568
#define STAT_N (NBR * COUT)                       // 512
#define CNT_PER_STAT (N_IMG * OHC * OWC)          // 200704
#define TOTAL_OUT ((size_t)N_IMG * NBR * COUT * PHO * PWO) // 25,690,112
#define CONV_ELEMS ((size_t)NBR * N_IMG * COUT * OHC * OWC)

// Compile-time A-fragment LDS offset for GEMM-K index k (clamped; k=147 pairs
// with the zeroed B row so the clamped A value contributes 0). Patch layout
// is [ci][row 0..8][col 0..36]; tile 1 (oh+1) adds ROW1_OFF.
__host__ __device__ constexpr int aoff(int k) {
    int kc = k > (KDIM - 1) ? (KDIM - 1) : k;
    int ci = kc / 49, rm = kc % 49;
    return (ci * PATCH_H + rm / 7) * PATCH_W + (rm % 7);
}

// ---------------------------------------------------------------------------
// Phase 0: zero the BN accumulation buffers (ws is poisoned by the harness).
// ---------------------------------------------------------------------------
__global__ void init_stats_kernel(float* __restrict__ stats) {
    int i = blockIdx.x * blockDim.x + threadIdx.x;
    if (i < 2 * STAT_N) stats[i] = 0.0f;
}

// ---------------------------------------------------------------------------
// Phase 1: implicit-GEMM 7x7/s2 conv via V_WMMA_F32_16X16X4_F32.
// One wave = 32 output pixels (rows oh, oh+1 x 16 cols) x 64 channels.
// B fragments are register-reused across both M tiles: 8 WMMA per 12 ds_load.
// ---------------------------------------------------------------------------
__global__ __launch_bounds__(128) void conv_wmma_kernel(
    const float* __restrict__ x,      // [N, B, Cin, 224, 224]
    const float* __restrict__ w,      // [B, Cout, Cin, 7, 7]
    float* __restrict__ conv,         // ws: [B, N, Cout, 112, 112]
    float* __restrict__ stat_sum,     // ws: [B*Cout]
    float* __restrict__ stat_sq)      // ws: [B*Cout]
{
    __shared__ float smem[WSH_FLOATS + NWAVES * PATCH_FLOATS]; // 54.5 KB
    float* wsh = smem;

    const int tid  = threadIdx.x;
    const int wave = tid >> 5;
    const int lane = tid & 31;

    const int b = blockIdx.x / BLOCKS_PER_BRANCH;
    const int t = (blockIdx.x % BLOCKS_PER_BRANCH) * NWAVES + wave;
    const int n   = t / (56 * 7);
    const int rem = t % (56 * 7);
    const int oh0 = (rem / 7) * 2;          // this wave does rows oh0, oh0+1
    const int ow0 = (rem % 7) * 16;

    // ---- stage weights transposed into LDS: wsh[k*65 + co] (global coalesced,
    //      LDS stride-65 -> conflict-free), plus a zero row at k = 147.
    for (int idx = tid; idx < COUT * KDIM; idx += 128) {
        int co = idx / KDIM, k = idx % KDIM;
        wsh[k * WSH_STRIDE + co] = w[(size_t)b * (COUT * KDIM) + idx];
    }
    if (tid < WSH_STRIDE) wsh[KDIM * WSH_STRIDE + tid] = 0.0f;

    // ---- stage shared input patch: [ci][r 0..8][c 0..36], halo zeroed
    float* patch = smem + WSH_FLOATS + wave * PATCH_FLOATS;
    const float* xb = x + (size_t)(n * NBR + b) * (CIN * HIN * WIN);
    for (int idx = lane; idx < PATCH_FLOATS; idx += 32) {
        int ci = idx / (PATCH_H * PATCH_W);
        int r2 = idx % (PATCH_H * PATCH_W);
        int r  = r2 / PATCH_W;
        int c  = r2 % PATCH_W;
        int ih = oh0 * 2 - 3 + r;
        int iw = ow0 * 2 - 3 + c;
        float v = 0.0f;
        if ((unsigned)ih < (unsigned)HIN && (unsigned)iw < (unsigned)WIN)
            v = xb[((size_t)ci * HIN + ih) * WIN + iw];
        patch[idx] = v;
    }
    __syncthreads();

    v8f acc0[4], acc1[4];
    #pragma unroll
    for (int j = 0; j < 4; ++j) { acc0[j] = (v8f){}; acc1[j] = (v8f){}; }

    const int m  = lane & 15;   // M row (pixel) for A, N col (channel) for B/C
    const int hi = lane >> 4;   // lanes 16-31 carry K+2, K+3

    const float* pA   = patch + 2 * m;
    const float* wshB = wsh + m + hi * (2 * WSH_STRIDE);

    #pragma unroll
    for (int k0 = 0; k0 < KDIM; k0 += 4) {
        const int offx = hi ? aoff(k0 + 2) : aoff(k0);      // literal cndmask
        const int offy = hi ? aoff(k0 + 3) : aoff(k0 + 1);
        v2f a0, a1;
        a0.x = pA[offx];
        a0.y = pA[offy];
        a1.x = pA[offx + ROW1_OFF];                         // oh+1: +2 rows
        a1.y = pA[offy + ROW1_OFF];

        #pragma unroll
        for (int j = 0; j < 4; ++j) {
            v2f bf;
            bf.x = wshB[k0 * WSH_STRIDE + 16 * j];          // ds imm offset
            bf.y = wshB[(k0 + 1) * WSH_STRIDE + 16 * j];    // ds imm offset
            acc0[j] = __builtin_amdgcn_wmma_f32_16x16x4_f32(
                false, a0, false, bf, (short)0, acc0[j], false, false);
            acc1[j] = __builtin_amdgcn_wmma_f32_16x16x4_f32(
                false, a1, false, bf, (short)0, acc1[j], false, false);
        }
    }

    // ---- BN statistics from registers (lanes l and l^16 share a channel)
    #pragma unroll
    for (int j = 0; j < 4; ++j) {
        const int co = 16 * j + m;
        float s = 0.0f, s2 = 0.0f;
        #pragma unroll
        for (int r = 0; r < 8; ++r) {
            float v0 = acc0[j][r], v1 = acc1[j][r];
            s  += v0 + v1;
            s2 += v0 * v0 + v1 * v1;
        }
        s  += __shfl_xor(s, 16, 32);
        s2 += __shfl_xor(s2, 16, 32);
        if (!hi) {
            atomicAdd(&stat_sum[b * COUT + co], s);
            atomicAdd(&stat_sq [b * COUT + co], s2);
        }
    }

    // ---- coalesced store: transpose both 16px x 64ch tiles through LDS
    __syncthreads();                        // everyone done reading wsh/patch
    float* trans = smem + wave * TR_FLOATS; // 4 * 2304 floats = 36.9 KB
    #pragma unroll
    for (int j = 0; j < 4; ++j) {
        const int co = 16 * j + m;
        #pragma unroll
        for (int r = 0; r < 8; ++r) {
            trans[co * TR_STRIDE + r + 8 * hi] = acc0[j][r];          // tile 0
            trans[(COUT + co) * TR_STRIDE + r + 8 * hi] = acc1[j][r]; // tile 1
        }
    }
    // 8 lanes cover one 64B row segment -> b64 stores, 4 lines per instruction
    const size_t cbase = (((size_t)(b * N_IMG + n) * COUT) * OHC + oh0) * OWC + ow0;
    const int coSub = lane >> 3;            // 0..3
    const int owl   = (lane & 7) * 2;       // 0,2,...,14
    #pragma unroll
    for (int it = 0; it < 16; ++it) {
        const int co = it * 4 + coSub;
        const float2 v0 = *reinterpret_cast<const float2*>(
            &trans[co * TR_STRIDE + owl]);
        const float2 v1 = *reinterpret_cast<const float2*>(
            &trans[(COUT + co) * TR_STRIDE + owl]);
        float* rowp = conv + cbase + (size_t)co * (OHC * OWC) + owl;
        *reinterpret_cast<float2*>(rowp) = v0;            // row oh0
        *reinterpret_cast<float2*>(rowp + OWC) = v1;      // row oh0+1
    }
}

// ---------------------------------------------------------------------------
// Phase 2: fold BN stats into per-channel scale/shift.
// ---------------------------------------------------------------------------
__global__ void finalize_stats_kernel(
    const float* __restrict__ stat_sum, const float* __restrict__ stat_sq,
    const float* __restrict__ gamma,    const float* __restrict__ beta,
    float* __restrict__ scaleB,         float* __restrict__ shiftB)
{
    int i = blockIdx.x * blockDim.x + threadIdx.x;
    if (i < STAT_N) {
        const float inv  = 1.0f / (float)CNT_PER_STAT;
        const float mean = stat_sum[i] * inv;
        const float var  = stat_sq[i] * inv - mean * mean;
        const float sc   = gamma[i] * rsqrtf(var + 1e-5f);
        scaleB[i] = sc;
        shiftB[i] = beta[i] - mean * sc;
    }
}

// ---------------------------------------------------------------------------
// Phase 3: affine + ReLU + 3x3/s2 maxpool (ReLU >= 0, so init max at 0 is exact).
// ---------------------------------------------------------------------------
__global__ __launch_bounds__(256) void pool_kernel(
    const float* __restrict__ conv, const float* __restrict__ scaleB,
    const float* __restrict__ shiftB, float* __restrict__ out)
{
    size_t idx = (size_t)blockIdx.x * blockDim.x + threadIdx.x;
    if (idx >= TOTAL_OUT) return;
    int pw = (int)(idx % PWO);  size_t r = idx / PWO;
    int ph = (int)(r % PHO);    r /= PHO;
    int co = (int)(r % COUT);   r /= COUT;
    int b  = (int)(r % NBR);
    int n  = (int)(r / NBR);

    const float sc = scaleB[b * COUT + co];
    const float sh = shiftB[b * COUT + co];
    const float* base = conv + (((size_t)(b * N_IMG + n) * COUT + co) * OHC) * OWC;

    float mx = 0.0f;
    #pragma unroll
    for (int dy = 0; dy < 3; ++dy) {
        int yh = 2 * ph - 1 + dy;
        if ((unsigned)yh >= (unsigned)OHC) continue;
        #pragma unroll
        for (int dx = 0; dx < 3; ++dx) {
            int xw = 2 * pw - 1 + dx;
            if ((unsigned)xw >= (unsigned)OWC) continue;
            float v = fmaf(sc, base[(size_t)yh * OWC + xw], sh);
            v = v > 0.0f ? v : 0.0f;
            mx = fmaxf(mx, v);
        }
    }
    out[idx] = mx;
}

// ---------------------------------------------------------------------------
extern "C" void kernel_launch(void* const* d_in, const int* in_sizes, int n_in,
                              void* d_out, int out_size, void* d_ws, size_t ws_size,
                              hipStream_t stream) {
    const float* x      = (const float*)d_in[0];
    const float* conv_w = (const float*)d_in[1];
    const float* gamma  = (const float*)d_in[2];
    const float* beta   = (const float*)d_in[3];
    float* out = (float*)d_out;

    // Workspace layout: [conv activations | sum | sumsq | scale | shift]
    float* conv     = (float*)d_ws;
    float* stat_sum = conv + CONV_ELEMS;
    float* stat_sq  = stat_sum + STAT_N;
    float* scaleB   = stat_sq + STAT_N;
    float* shiftB   = scaleB + STAT_N;

    init_stats_kernel<<<(2 * STAT_N + 255) / 256, 256, 0, stream>>>(stat_sum);

    conv_wmma_kernel<<<NBR * BLOCKS_PER_BRANCH, 128, 0, stream>>>(
        x, conv_w, conv, stat_sum, stat_sq);

    finalize_stats_kernel<<<(STAT_N + 255) / 256, 256, 0, stream>>>(
        stat_sum, stat_sq, gamma, beta, scaleB, shiftB);

    size_t total = TOTAL_OUT;
    pool_kernel<<<(unsigned)((total + 255) / 256), 256, 0, stream>>>(
        conv, scaleB, shiftB, out);
}